// RotateQuantizer_37228776521743
// MI455X (gfx1250) — compile-verified
//
#include <hip/hip_runtime.h>

// RotateQuantizer: out = (X @ H) @ H^T,  H = diag(s) * Had_4096 / 64.
// Two fast Walsh-Hadamard transforms per row; the 16-point butterfly stages
// run on the CDNA5 matrix cores via V_WMMA_F32_16X16X4_F32 (f32 precision,
// matching the reference numerics). Memory-bound: ~512 MiB total traffic.

typedef __attribute__((ext_vector_type(2))) float v2f;
typedef __attribute__((ext_vector_type(8))) float v8f;

#define DDIM 4096
// LDS layout for one 4096-float row: element (a,b,c) -> a*272 + b*17 + c
// (digits base 16: d = a*256 + b*16 + c). Odd stride 17 => conflict-free
// lane addressing for the strided WMMA fragment gathers.
#define LDS_FLOATS 4352

__device__ __forceinline__ int ldsIdx(int a, int b, int c) {
  return a * 272 + b * 17 + c;
}

// Had_16[k][n] = +1/-1 by parity of popcount(k & n) (Sylvester construction).
__device__ __forceinline__ float hadSign(int k, int n) {
  return (__popc(k & n) & 1) ? -1.0f : 1.0f;
}

// signs[d] = sign(rot_mat[d][0]) since rot_mat[d][0] = s[d]/64, Had[d][0] = +1.
__global__ void rq_extract_signs(const float* __restrict__ rot,
                                 float* __restrict__ signs) {
  int d = blockIdx.x * blockDim.x + threadIdx.x;
  if (d < DDIM) {
    signs[d] = (rot[(size_t)d * DDIM] > 0.0f) ? 1.0f : -1.0f;
  }
}

__device__ __forceinline__ v8f wmma_f32(v2f a, v2f b, v8f c) {
  return __builtin_amdgcn_wmma_f32_16x16x4_f32(
      /*neg_a=*/false, a, /*neg_b=*/false, b,
      /*c_mod=*/(short)0, c, /*reuse_a=*/false, /*reuse_b=*/false);
}

// One FWHT stage over the row held in LDS. Each wave owns tiles f=wave and
// f=wave+8; the two 16x16x16 products (4 chained 16x16x4 WMMAs each) are
// interleaved so the two accumulator chains hide the WMMA->WMMA RAW latency.
// stage 0: contract low digit  (c): baseMul=272, mulM=17, mulK=1
// stage 1: contract mid digit  (b): baseMul=272, mulM=1,  mulK=17
// stage 2: contract high digit (a): baseMul=17,  mulM=1,  mulK=272
// D(M,N) stores reuse mulK for N (in-place permuted write of the same
// 256-element tile; all loads precede stores in program order, LDS ops are
// in-order per wave, and tiles are disjoint across waves).
__device__ __forceinline__ void fwht_stage(float* sm, int baseMul, int mulM,
                                           int mulK, const v2f* bfrag,
                                           int lane, int wave) {
  const int m  = lane & 15;               // A row (M) for this lane
  const int n  = lane & 15;               // D column (N) for this lane
  const int kh = (lane & 16) ? 2 : 0;     // half-wave K offset (f32 A layout)
  const int mb = (lane & 16) ? 8 : 0;     // half-wave M offset (f32 C/D layout)

  const int base0 = wave * baseMul;
  const int base1 = (wave + 8) * baseMul;
  const int arow0 = base0 + m * mulM;
  const int arow1 = base1 + m * mulM;

  v8f acc0 = {}, acc1 = {};
#pragma unroll
  for (int ch = 0; ch < 4; ++ch) {
    const int k0 = ch * 4 + kh;
    v2f a0, a1;
    a0.x = sm[arow0 + (k0)     * mulK];
    a0.y = sm[arow0 + (k0 + 1) * mulK];
    a1.x = sm[arow1 + (k0)     * mulK];
    a1.y = sm[arow1 + (k0 + 1) * mulK];
    acc0 = wmma_f32(a0, bfrag[ch], acc0);  // two independent WMMA chains
    acc1 = wmma_f32(a1, bfrag[ch], acc1);
  }
#pragma unroll
  for (int v = 0; v < 8; ++v) {
    sm[base0 + (v + mb) * mulM + n * mulK] = acc0[v];
    sm[base1 + (v + mb) * mulM + n * mulK] = acc1[v];
  }
}

__global__ __launch_bounds__(256) void rq_fwht_kernel(
    const float* __restrict__ x, const float* __restrict__ signs,
    float* __restrict__ out) {
  __shared__ float sm[LDS_FLOATS];
  const int row  = blockIdx.x;
  const int t    = threadIdx.x;
  const int lane = t & 31;   // wave32 (gfx1250)
  const int wave = t >> 5;   // 8 waves per block

  // Precompute the H16 B-fragments once: ±1 (and ±1/64 for the normalized
  // stages). Held in 16 VGPRs for the whole kernel — no per-stage VALU churn.
  const int n  = lane & 15;
  const int kh = (lane & 16) ? 2 : 0;
  v2f bfrag1[4], bfragS[4];
#pragma unroll
  for (int ch = 0; ch < 4; ++ch) {
    const int k0 = ch * 4 + kh;
    bfrag1[ch].x = hadSign(k0, n);
    bfrag1[ch].y = hadSign(k0 + 1, n);
    bfragS[ch].x = bfrag1[ch].x * (1.0f / 64.0f);
    bfragS[ch].y = bfrag1[ch].y * (1.0f / 64.0f);
  }

  const float* xr = x + (size_t)row * DDIM;

  // Load row (b128 coalesced), apply input sign diagonal, scatter to LDS.
#pragma unroll
  for (int i = 0; i < 4; ++i) {
    const int d0 = t * 4 + i * 1024;
    const float4 xv = *(const float4*)(xr + d0);
    const float4 sv = *(const float4*)(signs + d0);
    const int a = d0 >> 8, b = (d0 >> 4) & 15, c = d0 & 15;
    float* p = &sm[ldsIdx(a, b, c)];
    p[0] = xv.x * sv.x;
    p[1] = xv.y * sv.y;
    p[2] = xv.z * sv.z;
    p[3] = xv.w * sv.w;
  }
  __syncthreads();

  // rotate:    y = (x*s) . Had / 64
  fwht_stage(sm, 272, 17, 1,   bfrag1, lane, wave); __syncthreads();
  fwht_stage(sm, 272, 1,  17,  bfrag1, lane, wave); __syncthreads();
  fwht_stage(sm, 17,  1,  272, bfragS, lane, wave); __syncthreads();
  // de-rotate: out = (y . Had / 64) * s
  fwht_stage(sm, 272, 17, 1,   bfrag1, lane, wave); __syncthreads();
  fwht_stage(sm, 272, 1,  17,  bfrag1, lane, wave); __syncthreads();
  fwht_stage(sm, 17,  1,  272, bfragS, lane, wave); __syncthreads();

  float* outr = out + (size_t)row * DDIM;
#pragma unroll
  for (int i = 0; i < 4; ++i) {
    const int d0 = t * 4 + i * 1024;
    const float4 sv = *(const float4*)(signs + d0);
    const int a = d0 >> 8, b = (d0 >> 4) & 15, c = d0 & 15;
    const float* p = &sm[ldsIdx(a, b, c)];
    float4 ov;
    ov.x = p[0] * sv.x;
    ov.y = p[1] * sv.y;
    ov.z = p[2] * sv.z;
    ov.w = p[3] * sv.w;
    *(float4*)(outr + d0) = ov;
  }
}

extern "C" void kernel_launch(void* const* d_in, const int* in_sizes, int n_in,
                              void* d_out, int out_size, void* d_ws, size_t ws_size,
                              hipStream_t stream) {
  const float* x   = (const float*)d_in[0];  // hidden_states [B,S,D] f32
  const float* rot = (const float*)d_in[1];  // rot_mat [D,D] f32
  float* out   = (float*)d_out;
  float* signs = (float*)d_ws;               // 4096 floats of scratch

  const int rows = in_sizes[0] / DDIM;       // B*S = 16384

  rq_extract_signs<<<DDIM / 256, 256, 0, stream>>>(rot, signs);
  rq_fwht_kernel<<<rows, 256, 0, stream>>>(x, signs, out);
}